// QPNet_41369124995342
// MI455X (gfx1250) — compile-verified
//
#include <hip/hip_runtime.h>
#include <math.h>

// ---------------------------------------------------------------------------
// QPNet on MI455X (gfx1250, wave32).
//   n = NVAR = 32 == wave size  ->  one wave per batch element for the IPM.
//   GEMMs use V_WMMA_F32_16X16X4_F32 (exact fp32 semantics).
//   Per-element SPD solve: LDS-resident 32x32 Cholesky, stride 33 (bank-safe).
// ---------------------------------------------------------------------------

#define NVAR   32
#define PAD    33          // 33 coprime with 64 banks -> conflict-free rows/cols
#define BATCH  16384
#define INF    __builtin_inff()

typedef float v2f __attribute__((ext_vector_type(2)));
typedef float v8f __attribute__((ext_vector_type(8)));

// ---------------- wave-level helpers (wave32) ------------------------------
__device__ __forceinline__ float wsum32(float v) {
    v += __shfl_xor(v, 16, 32);
    v += __shfl_xor(v, 8, 32);
    v += __shfl_xor(v, 4, 32);
    v += __shfl_xor(v, 2, 32);
    v += __shfl_xor(v, 1, 32);
    return v;
}
__device__ __forceinline__ float wmin32(float v) {
    v = fminf(v, __shfl_xor(v, 16, 32));
    v = fminf(v, __shfl_xor(v, 8, 32));
    v = fminf(v, __shfl_xor(v, 4, 32));
    v = fminf(v, __shfl_xor(v, 2, 32));
    v = fminf(v, __shfl_xor(v, 1, 32));
    return v;
}

// r_i = sum_j M[i][j] * v_j   (M in LDS, padded stride PAD; v lane-distributed)
__device__ __forceinline__ float matvec33(const float* M, float v, int lane) {
    float r = 0.f;
#pragma unroll
    for (int j = 0; j < 32; ++j)
        r = fmaf(M[lane * PAD + j], __shfl(v, j, 32), r);
    return r;
}

// In-place Cholesky of 32x32 SPD matrix in LDS (lower triangle), one wave.
__device__ __forceinline__ void chol32(float* S, int lane) {
    for (int k = 0; k < 32; ++k) {
        float dkk = S[k * PAD + k];          // broadcast read
        float lkk = sqrtf(dkk);
        float il  = 1.0f / lkk;
        if (lane == k)      S[k * PAD + k]    = lkk;
        else if (lane > k)  S[lane * PAD + k] *= il;   // column k, bank-safe
        __builtin_amdgcn_wave_barrier();
        float Sik = (lane > k) ? S[lane * PAD + k] : 0.f;
        for (int j = k + 1; j < 32; ++j) {
            float Ljk = S[j * PAD + k];       // broadcast read
            if (lane >= j) S[lane * PAD + j] -= Sik * Ljk;  // column j write
        }
        __builtin_amdgcn_wave_barrier();
    }
}

// Solve (L L^T) w = b with L = chol factor in LDS; b,w lane-distributed.
__device__ __forceinline__ float ltlsolve(const float* S, float b, int lane) {
    float r = b;
    for (int k = 0; k < 32; ++k) {           // forward: L y = b
        float yk = __shfl(r, k, 32) / S[k * PAD + k];
        if (lane == k)     r = yk;
        else if (lane > k) r = fmaf(-S[lane * PAD + k], yk, r);
    }
    for (int k = 31; k >= 0; --k) {          // backward: L^T w = y
        float wk = __shfl(r, k, 32) / S[k * PAD + k];
        if (lane == k)     r = wk;
        else if (lane < k) r = fmaf(-S[k * PAD + lane], wk, r);
    }
    return r;
}

// _get_step: min over lanes of (dv<0 ? -v/dv : inf)
__device__ __forceinline__ float stepmin(float v, float dv) {
    float s = (dv < 0.f) ? (-v / dv) : INF;
    return wmin32(s);
}

// ---------------- kernel 1: Q = 1e-3 I + H^T H, chol(Q), invQ --------------
__global__ void qp_prep_kernel(const float* __restrict__ H,
                               float* __restrict__ Qg,
                               float* __restrict__ invQg) {
    __shared__ float Hs[32 * 32];
    __shared__ float Ls[32 * PAD];
    __shared__ float Xs[32 * PAD];
    const int lane = threadIdx.x;            // launched with 32 threads

    for (int k = 0; k < 32; ++k) Hs[k * 32 + lane] = H[k * 32 + lane];
    __builtin_amdgcn_wave_barrier();

    // lane owns column j = lane of Q
    for (int i = 0; i < 32; ++i) {
        float q = (i == lane) ? 1e-3f : 0.f;
        for (int k = 0; k < 32; ++k)
            q = fmaf(Hs[k * 32 + i], Hs[k * 32 + lane], q);
        Qg[i * 32 + lane] = q;
        Ls[i * PAD + lane] = q;
    }
    __builtin_amdgcn_wave_barrier();

    chol32(Ls, lane);

    // invQ column j = lane: solve L L^T x = e_j (each lane independent column)
    for (int i = 0; i < 32; ++i) {
        float acc = (i == lane) ? 1.f : 0.f;
        for (int m = 0; m < i; ++m)
            acc = fmaf(-Ls[i * PAD + m], Xs[m * PAD + lane], acc);
        Xs[i * PAD + lane] = acc / Ls[i * PAD + i];
    }
    for (int i = 31; i >= 0; --i) {
        float acc = Xs[i * PAD + lane];
        for (int m = i + 1; m < 32; ++m)
            acc = fmaf(-Ls[m * PAD + i], Xs[m * PAD + lane], acc);
        Xs[i * PAD + lane] = acc / Ls[i * PAD + i];
    }
    for (int i = 0; i < 32; ++i) invQg[i * 32 + lane] = Xs[i * PAD + lane];
}

// ---------------- kernel 2: p = x @ W_in^T + b_in via WMMA fp32 ------------
// M=16384, K=64, N=32. One wave per 16x16 C tile; K-loop of 16x k=4 steps.
__global__ __launch_bounds__(128) void qp_gemm_in_kernel(
    const float* __restrict__ X, const float* __restrict__ Win,
    const float* __restrict__ bin, float* __restrict__ P) {
    const int lane = threadIdx.x & 31;
    const int wave = (blockIdx.x * 128 + threadIdx.x) >> 5;  // 0..2047
    const int m0 = (wave >> 1) * 16;
    const int n0 = (wave & 1) * 16;
    const int mrow = m0 + (lane & 15);
    const int ncol = n0 + (lane & 15);
    const int kh = lane >> 4;            // lanes 16-31 hold K+2 / M+8 halves

    v8f c = {};
#pragma unroll
    for (int k0 = 0; k0 < 64; k0 += 4) {
        // A: 16x4 fp32 tile of x  (lane row, VGPR pair = K{0,1} / K{2,3})
        v2f a = *(const v2f*)(X + mrow * 64 + k0 + 2 * kh);
        // B: 4x16 tile, B[k][n] = Win[n][k]
        v2f b = *(const v2f*)(Win + ncol * 64 + k0 + 2 * kh);
        c = __builtin_amdgcn_wmma_f32_16x16x4_f32(false, a, false, b,
                                                  (short)0, c, false, false);
    }
    float bn = bin[ncol];
#pragma unroll
    for (int r = 0; r < 8; ++r) {
        int row = m0 + r + 8 * kh;       // C layout: VGPR r = M rows r / r+8
        P[row * 32 + ncol] = c[r] + bn;
    }
}

// ---------------- kernel 3: per-element interior-point QP ------------------
// One wave (32 lanes) per batch element; lane = variable index.
__global__ __launch_bounds__(256) void qp_solve_kernel(
    float* P,                      // in: p ; out: best_x (written at the end)
    const float* __restrict__ Qg, const float* __restrict__ invQg) {
    __shared__ float Qs[32 * PAD];
    __shared__ float iQs[32 * PAD];
    __shared__ float Ss[8 * 32 * PAD];       // per-wave S slab
    const int lane = threadIdx.x & 31;
    const int wave = threadIdx.x >> 5;

    for (int t = threadIdx.x; t < 1024; t += 256) {
        int i = t >> 5, j = t & 31;
        Qs[i * PAD + j]  = Qg[t];
        iQs[i * PAD + j] = invQg[t];
    }
    __syncthreads();

    float* S = &Ss[wave * 32 * PAD];
    const int b = blockIdx.x * 8 + wave;
    const float p = P[b * 32 + lane];

    // ---- initial kkt solve: d = 1, rx = p, rs = rz = 0; S = invQ + I ----
#pragma unroll
    for (int j = 0; j < 32; ++j)
        S[lane * PAD + j] = iQs[lane * PAD + j] + ((j == lane) ? 1.f : 0.f);
    __builtin_amdgcn_wave_barrier();
    chol32(S, lane);

    float iqp = matvec33(iQs, p, lane);
    float w0  = -ltlsolve(S, -iqp, lane);
    float xk  = matvec33(iQs, -p + w0, lane);
    float s   = -w0;
    float z   =  w0;
    { float m = wmin32(s); if (m < 0.f) s = s - m + 1.f; }   // _shift_ge_one
    { float m = wmin32(z); if (m < 0.f) z = z - m + 1.f; }

    float best_x = xk;
    float best_r = INF;

    for (int it = 0; it < 5; ++it) {
        float rx = -z + matvec33(Qs, xk, lane) + p;
        float rs = z;
        float rz = -xk + s;
        float sz = wsum32(s * z);
        float mu = fabsf(sz) * (1.0f / 32.0f);
        float resid = sqrtf(wsum32(rz * rz)) + sqrtf(wsum32(rx * rx)) + 32.f * mu;
        if (resid < best_r) { best_r = resid; best_x = xk; }

        float d  = z / s;
        float id = 1.0f / d;

        // S = invQ + diag(1/d); one factorization serves both KKT solves
#pragma unroll
        for (int j = 0; j < 32; ++j)
            S[lane * PAD + j] = iQs[lane * PAD + j] + ((j == lane) ? id : 0.f);
        __builtin_amdgcn_wave_barrier();
        chol32(S, lane);

        // affine step
        float iqrx = matvec33(iQs, rx, lane);
        float hvec = -iqrx + rs * id - rz;
        float wa   = -ltlsolve(S, hvec, lane);
        float dxa  = matvec33(iQs, -rx + wa, lane);
        float dsa  = (-rs - wa) * id;
        float dza  = wa;

        float alpha = fminf(fminf(stepmin(z, dza), stepmin(s, dsa)), 1.0f);
        float t1 = s + alpha * dsa, t2 = z + alpha * dza;
        float ratio = wsum32(t1 * t2) / sz;
        float sig = ratio * ratio * ratio;
        float rsc = (fmaf(-mu, sig, dsa * dza)) / s;

        // corrector step (rx = rz = 0)
        float wc  = -ltlsolve(S, rsc * id, lane);
        float dxc = matvec33(iQs, wc, lane);
        float dsc = (-rsc - wc) * id;
        float dzc = wc;

        float dx = dxa + dxc, ds = dsa + dsc, dz = dza + dzc;
        alpha = fminf(0.999f * fminf(stepmin(z, dz), stepmin(s, ds)), 1.0f);
        xk = fmaf(alpha, dx, xk);
        s  = fmaf(alpha, ds, s);
        z  = fmaf(alpha, dz, z);
    }

    // overwrite p slot with best_x (p no longer needed)
    P[b * 32 + lane] = best_x;
}

// ---------------- kernel 4: out = clip(best_x @ W_out^T + b_out) -----------
// M=16384, K=32, N=16 via WMMA fp32.
__global__ __launch_bounds__(128) void qp_gemm_out_kernel(
    const float* __restrict__ BX, const float* __restrict__ Wout,
    const float* __restrict__ bout, float* __restrict__ Out) {
    const int lane = threadIdx.x & 31;
    const int wave = (blockIdx.x * 128 + threadIdx.x) >> 5;  // 0..1023
    const int m0 = wave * 16;
    const int mrow = m0 + (lane & 15);
    const int ncol = lane & 15;
    const int kh = lane >> 4;

    v8f c = {};
#pragma unroll
    for (int k0 = 0; k0 < 32; k0 += 4) {
        v2f a = *(const v2f*)(BX + mrow * 32 + k0 + 2 * kh);
        v2f b = *(const v2f*)(Wout + ncol * 32 + k0 + 2 * kh);
        c = __builtin_amdgcn_wmma_f32_16x16x4_f32(false, a, false, b,
                                                  (short)0, c, false, false);
    }
    float bo = bout[ncol];
#pragma unroll
    for (int r = 0; r < 8; ++r) {
        int row = m0 + r + 8 * kh;
        float v = c[r] + bo;
        Out[row * 16 + ncol] = fminf(fmaxf(v, 0.f), 1.f);
    }
}

// ---------------------------------------------------------------------------
extern "C" void kernel_launch(void* const* d_in, const int* in_sizes, int n_in,
                              void* d_out, int out_size, void* d_ws, size_t ws_size,
                              hipStream_t stream) {
    (void)in_sizes; (void)n_in; (void)out_size; (void)ws_size;
    const float* x    = (const float*)d_in[0];   // 16384 x 64
    const float* H    = (const float*)d_in[1];   // 32 x 32
    const float* Win  = (const float*)d_in[2];   // 32 x 64
    const float* bin  = (const float*)d_in[3];   // 32
    const float* Wout = (const float*)d_in[4];   // 16 x 32
    const float* bout = (const float*)d_in[5];   // 16
    float* out = (float*)d_out;                  // 16384 x 16

    float* ws    = (float*)d_ws;
    float* Qg    = ws;            // 1024
    float* invQg = ws + 1024;     // 1024
    float* P     = ws + 2048;     // 16384*32 (p, then best_x)

    qp_prep_kernel<<<1, 32, 0, stream>>>(H, Qg, invQg);
    qp_gemm_in_kernel<<<512, 128, 0, stream>>>(x, Win, bin, P);
    qp_solve_kernel<<<2048, 256, 0, stream>>>(P, Qg, invQg);
    qp_gemm_out_kernel<<<256, 128, 0, stream>>>(P, Wout, bout, out);
}